// GAT_4260607558135
// MI455X (gfx1250) — compile-verified
//
#include <hip/hip_runtime.h>
#include <hip/hip_bf16.h>

// ---------------------------------------------------------------------------
// GAT (3-layer, edge-featured) for MI455X / gfx1250.
// GEMMs use V_WMMA_F32_16X16X4_F32 (f32 matrix pipe, full precision).
// Edge softmax/scatter is the HBM-bound part: monotonic-uint atomic max,
// f32 atomic adds.
// ---------------------------------------------------------------------------

#define N_NODES 50000
#define N_EDGES 500000
#define HC      256      // H * C = 8 * 32 (hidden width, all layers)
#define HEADS   8
#define CH      32       // per-head channels
#define EDIM    32
#define NEG_SLOPE 0.2f
#define SOFT_EPS  1e-16f

typedef float v2f __attribute__((ext_vector_type(2)));
typedef float v8f __attribute__((ext_vector_type(8)));

// ---------------- monotonic float <-> uint mapping for atomic max ----------
__device__ __forceinline__ unsigned enc_f32(float f) {
    unsigned u = __float_as_uint(f);
    return (f >= 0.0f) ? (u | 0x80000000u) : ~u;
}
__device__ __forceinline__ float dec_f32(unsigned u) {
    return (u & 0x80000000u) ? __uint_as_float(u & 0x7FFFFFFFu)
                             : __uint_as_float(~u);
}

// ---------------------------------------------------------------------------
// Generic f32 WMMA GEMM:  D[M,Ncol] = A[M,K] @ B[K,Ncol] (+ Cin) (+biases)(ELU)
// One wave per 16x16 tile of D; K stepped by 4 via v_wmma_f32_16x16x4_f32.
//   A fragment: lane L holds A[tm + (L&15)][k + 2*(L>>4) + {0,1}]  (float2)
//   B fragment: lane L holds B[k + 2*(L>>4) + {0,1}][tn + (L&15)]
//   C/D       : lane L, vgpr r holds D[tm + 8*(L>>4) + r][tn + (L&15)]
// In-place D == Cin is safe: each tile read/written by exactly one wave.
// ---------------------------------------------------------------------------
__global__ void __launch_bounds__(128)
gemm_f32_wmma(const float* __restrict__ A, int lda,
              const float* __restrict__ B, int ldb,
              const float* __restrict__ Cin, int ldc,
              float* __restrict__ D, int ldd,
              int M, int Ncol, int K,
              const float* __restrict__ bias0,
              const float* __restrict__ bias1,
              int apply_elu)
{
    const int lane    = threadIdx.x & 31;
    const int wave    = threadIdx.x >> 5;
    const int tiles_n = Ncol >> 4;
    const int tile    = blockIdx.x * 4 + wave;
    if (tile >= (M >> 4) * tiles_n) return;

    const int tm    = (tile / tiles_n) << 4;
    const int tn    = (tile % tiles_n) << 4;
    const int r15   = lane & 15;
    const int khalf = (lane >> 4) << 1;          // 0 or 2
    const int arow  = tm + r15;
    const int bcol  = tn + r15;
    const int crow0 = tm + ((lane >> 4) << 3);

    v8f c = {0.f, 0.f, 0.f, 0.f, 0.f, 0.f, 0.f, 0.f};
    if (Cin) {
#pragma unroll
        for (int r = 0; r < 8; ++r)
            c[r] = Cin[(size_t)(crow0 + r) * ldc + bcol];
    }

    const float* Aptr = A + (size_t)arow * lda + khalf;
#pragma unroll 4
    for (int k = 0; k < K; k += 4) {
        v2f a = *(const v2f*)(Aptr + k);                  // global_load_b64
        const float* Bp = B + (size_t)(k + khalf) * ldb + bcol;
        v2f b;
        b.x = Bp[0];
        b.y = Bp[ldb];
        c = __builtin_amdgcn_wmma_f32_16x16x4_f32(
                false, a, false, b, (short)0, c, false, false);
    }

    float badd = 0.0f;
    if (bias0) badd += bias0[bcol];
    if (bias1) badd += bias1[bcol];
#pragma unroll
    for (int r = 0; r < 8; ++r) {
        float v = c[r] + badd;
        if (apply_elu) v = (v > 0.0f) ? v : (__expf(v) - 1.0f);
        D[(size_t)(crow0 + r) * ldd + bcol] = v;
    }
}

// --------------- per-node attention coefficients a_src / a_dst -------------
__global__ void __launch_bounds__(256)
attn_node(const float* __restrict__ h,
          const float* __restrict__ att_src,
          const float* __restrict__ att_dst,
          float* __restrict__ a_src, float* __restrict__ a_dst, int n)
{
    int i = blockIdx.x * blockDim.x + threadIdx.x;
    if (i >= n * HEADS) return;
    int node = i >> 3, head = i & 7;
    const float* hp = h + (size_t)node * HC + head * CH;
    const float* as = att_src + head * CH;
    const float* ad = att_dst + head * CH;
    float s = 0.0f, d = 0.0f;
#pragma unroll
    for (int cc = 0; cc < CH; ++cc) {
        float v = hp[cc];
        s += v * as[cc];
        d += v * ad[cc];
    }
    a_src[i] = s;
    a_dst[i] = d;
}

// --------------- Ve[d][head] = sum_c We[d][head*32+c] * att_edge[head][c] --
__global__ void __launch_bounds__(256)
compute_ve(const float* __restrict__ We, const float* __restrict__ att_edge,
           float* __restrict__ Ve)
{
    int t = threadIdx.x;             // 256 = EDIM * HEADS
    int d = t >> 3, head = t & 7;
    float s = 0.0f;
#pragma unroll
    for (int cc = 0; cc < CH; ++cc)
        s += We[(size_t)d * HC + head * CH + cc] * att_edge[head * CH + cc];
    Ve[t] = s;                       // Ve[d*8 + head]
}

// --------------- a_edge = edge_attr @ Ve  ([E,8], K=32) --------------------
__global__ void __launch_bounds__(256)
edge_aedge(const float* __restrict__ edge_attr, const float* __restrict__ Ve,
           float* __restrict__ aedge, int ne)
{
    int i = blockIdx.x * blockDim.x + threadIdx.x;
    if (i >= ne * HEADS) return;
    int e = i >> 3, head = i & 7;
    const float* ea = edge_attr + (size_t)e * EDIM;
    float s = 0.0f;
#pragma unroll
    for (int d = 0; d < EDIM; ++d)
        s += ea[d] * Ve[d * HEADS + head];
    aedge[i] = s;
}

// --------------- alpha_raw + leaky-relu + segment max ----------------------
__global__ void __launch_bounds__(256)
edge_alpha_max(const int* __restrict__ src, const int* __restrict__ dst,
               const float* __restrict__ a_src, const float* __restrict__ a_dst,
               float* __restrict__ alpha, unsigned* __restrict__ amax_u, int ne)
{
    int i = blockIdx.x * blockDim.x + threadIdx.x;
    if (i >= ne * HEADS) return;
    int e = i >> 3, head = i & 7;
    int s = src[e], t = dst[e];
    float a = a_src[s * HEADS + head] + a_dst[t * HEADS + head] + alpha[i];
    a = (a > 0.0f) ? a : NEG_SLOPE * a;
    alpha[i] = a;
    atomicMax(&amax_u[t * HEADS + head], enc_f32(a));
}

// --------------- exp(alpha - amax[dst]) + segment sum ----------------------
__global__ void __launch_bounds__(256)
edge_exp_denom(const int* __restrict__ dst, float* __restrict__ alpha,
               const unsigned* __restrict__ amax_u, float* __restrict__ denom,
               int ne)
{
    int i = blockIdx.x * blockDim.x + threadIdx.x;
    if (i >= ne * HEADS) return;
    int e = i >> 3, head = i & 7;
    int t = dst[e];
    float m = dec_f32(amax_u[t * HEADS + head]);
    float v = __expf(alpha[i] - m);
    alpha[i] = v;
    atomicAdd(&denom[t * HEADS + head], v);
}

// --------------- message scatter: accum[dst] += h[src] * alpha/denom -------
// one wave per edge; each lane owns 8 contiguous channels (one head / 4 lanes)
__global__ void __launch_bounds__(256)
edge_scatter(const int* __restrict__ src, const int* __restrict__ dst,
             const float* __restrict__ h, const float* __restrict__ alpha,
             const float* __restrict__ denom, float* __restrict__ accum, int ne)
{
    int e = blockIdx.x * 8 + (threadIdx.x >> 5);
    if (e >= ne) return;
    int lane = threadIdx.x & 31;
    int head = lane >> 2;
    int s = src[e], t = dst[e];
    float w = alpha[(size_t)e * HEADS + head] /
              (denom[(size_t)t * HEADS + head] + SOFT_EPS);
    const float* hp = h + (size_t)s * HC + lane * 8;
    float*       op = accum + (size_t)t * HC + lane * 8;
#pragma unroll
    for (int j = 0; j < 8; ++j)
        atomicAdd(op + j, hp[j] * w);
}

// --------------- final layer: mean over heads + bias (into cols 0..31) -----
__global__ void __launch_bounds__(256)
mean_bias(float* __restrict__ accum, const float* __restrict__ bias, int n)
{
    int i = blockIdx.x * blockDim.x + threadIdx.x;
    if (i >= n * CH) return;
    int node = i >> 5, cc = i & 31;
    float s = 0.0f;
#pragma unroll
    for (int head = 0; head < HEADS; ++head)
        s += accum[(size_t)node * HC + head * CH + cc];
    accum[(size_t)node * HC + cc] = s * 0.125f + bias[cc];
}

// ---------------------------------------------------------------------------
extern "C" void kernel_launch(void* const* d_in, const int* in_sizes, int n_in,
                              void* d_out, int out_size, void* d_ws, size_t ws_size,
                              hipStream_t stream)
{
    (void)in_sizes; (void)n_in; (void)out_size; (void)ws_size;
    const int N = N_NODES, E = N_EDGES;

    const float* x0        = (const float*)d_in[0];
    const int*   eidx      = (const int*)d_in[1];
    const int*   srcv      = eidx;
    const int*   dstv      = eidx + E;
    const float* edge_attr = (const float*)d_in[2];

    // ---- workspace carve-up (floats) ----
    float* ws     = (float*)d_ws;
    float* xA     = ws;                               // N*256
    float* hbuf   = xA    + (size_t)N * HC;           // N*256
    float* accum  = hbuf  + (size_t)N * HC;           // N*256
    float* a_src  = accum + (size_t)N * HC;           // N*8
    float* a_dst  = a_src + (size_t)N * HEADS;        // N*8
    float* denom  = a_dst + (size_t)N * HEADS;        // N*8
    unsigned* amax_u = (unsigned*)(denom + (size_t)N * HEADS);  // N*8
    float* alpha  = (float*)(amax_u + (size_t)N * HEADS);       // E*8
    float* Ve     = alpha + (size_t)E * HEADS;        // 256

    const int tilesHC  = (N / 16) * (HC / 16);        // full-width GEMM tiles
    const int gB       = 256;
    const int gEH      = (E * HEADS + gB - 1) / gB;
    const int gNH      = (N * HEADS + gB - 1) / gB;

    for (int layer = 0; layer < 3; ++layer) {
        const int base = 3 + 8 * layer;
        const float* W        = (const float*)d_in[base + 0];
        const float* att_src  = (const float*)d_in[base + 1];
        const float* att_dst  = (const float*)d_in[base + 2];
        const float* We       = (const float*)d_in[base + 3];
        const float* att_edge = (const float*)d_in[base + 4];
        const float* bias     = (const float*)d_in[base + 5];
        const float* skipW    = (const float*)d_in[base + 6];
        const float* skipb    = (const float*)d_in[base + 7];

        // buffer rotation (keeps every kernel read-set disjoint from its
        // cross-wave write-set; in-place skip-GEMM is per-tile safe):
        //   L0: xin=x0    scat=accum  xout=xA
        //   L1: xin=xA    scat=accum  xout=accum (in-place D==Cin)
        //   L2: xin=accum scat=xA     xout=d_out
        const float* xin  = (layer == 0) ? x0 : (layer == 1 ? xA : accum);
        float*       scat = (layer == 2) ? xA : accum;

        hipMemsetAsync(scat,   0, (size_t)N * HC    * sizeof(float), stream);
        hipMemsetAsync(denom,  0, (size_t)N * HEADS * sizeof(float), stream);
        hipMemsetAsync(amax_u, 0, (size_t)N * HEADS * sizeof(unsigned), stream);

        // h = x @ W   [N,256] x [256,256]
        gemm_f32_wmma<<<(tilesHC + 3) / 4, 128, 0, stream>>>(
            xin, HC, W, HC, nullptr, 0, hbuf, HC, N, HC, HC,
            nullptr, nullptr, 0);

        attn_node<<<gNH, gB, 0, stream>>>(hbuf, att_src, att_dst,
                                          a_src, a_dst, N);
        compute_ve<<<1, 256, 0, stream>>>(We, att_edge, Ve);
        edge_aedge<<<gEH, gB, 0, stream>>>(edge_attr, Ve, alpha, E);
        edge_alpha_max<<<gEH, gB, 0, stream>>>(srcv, dstv, a_src, a_dst,
                                               alpha, amax_u, E);
        edge_exp_denom<<<gEH, gB, 0, stream>>>(dstv, alpha, amax_u, denom, E);
        edge_scatter<<<(E + 7) / 8, 256, 0, stream>>>(srcv, dstv, hbuf,
                                                      alpha, denom, scat, E);

        if (layer < 2) {
            // xout = ELU( scat + bias + skipb + xin @ skipW )
            float* xout = (layer == 0) ? xA : accum;
            gemm_f32_wmma<<<(tilesHC + 3) / 4, 128, 0, stream>>>(
                xin, HC, skipW, HC, scat, HC, xout, HC, N, HC, HC,
                bias, skipb, 1);
        } else {
            // mean over heads + bias into scat[:, 0:32], then
            // d_out = scat[:,0:32] + skipb + xin @ skipW   [N,32]
            mean_bias<<<(N * CH + gB - 1) / gB, gB, 0, stream>>>(scat, bias, N);
            const int tiles2 = (N / 16) * (CH / 16);
            gemm_f32_wmma<<<(tiles2 + 3) / 4, 128, 0, stream>>>(
                xin, HC, skipW, CH, scat, HC, (float*)d_out, CH, N, CH, HC,
                nullptr, skipb, 0);
        }
    }
}